// LinearLayer_34823594836412
// MI455X (gfx1250) — compile-verified
//
#include <hip/hip_runtime.h>

typedef float v2f __attribute__((ext_vector_type(2)));
typedef float v4f __attribute__((ext_vector_type(4)));
typedef float v8f __attribute__((ext_vector_type(8)));
typedef int   v4i __attribute__((ext_vector_type(4)));

#define NTHREADS 256
#define TILE 16            // nodes per tile (per pipeline stage)
#define ROW 512            // concatenated irrep row length (32*(1+3+5+7))
#define XSTRIDE 516        // padded LDS row stride -> conflict-free A reads
#define WKSTRIDE 40        // ldsW k-stride: 2-k half offset = 80 words = 16 banks (disjoint halves)
#define WSSTRIDE (32 * WKSTRIDE)
#define NBLOCKS 4096

#if defined(__AMDGCN__) && __has_builtin(__builtin_amdgcn_global_load_async_to_lds_b128)
#define HAVE_ASYNC 1
#pragma message("CDNA5: using global_load_async_to_lds_b128 async staging path")
#else
#define HAVE_ASYNC 0
#pragma message("CDNA5: async-to-LDS builtin not found; using sync LDS copy fallback")
#endif

__device__ __forceinline__ void issue_chunk(const float* gsrc, float* ldst) {
#if HAVE_ASYNC
    __builtin_amdgcn_global_load_async_to_lds_b128(
        (__attribute__((address_space(1))) v4i*)(void*)const_cast<float*>(gsrc),
        (__attribute__((address_space(3))) v4i*)(void*)ldst,
        /*offset=*/0, /*cpol=*/0);
#else
    *(v4f*)ldst = *(const v4f*)gsrc;
#endif
}

template <int CNT>
__device__ __forceinline__ void wait_async_le() {
#if HAVE_ASYNC
#if __has_builtin(__builtin_amdgcn_s_wait_asynccnt)
    __builtin_amdgcn_s_wait_asynccnt(CNT);
#else
    asm volatile("s_wait_asynccnt %0" :: "i"(CNT) : "memory");
#endif
#endif
}

// Map combined chunk index j (0..2047, b128 units of one 16x512 tile) -> (l, row, col)
// Chunk counts per l: l0:128, l1:384, l2:640, l3:896 (boundaries 128/512/1152/2048).
__device__ __forceinline__ void map_chunk(int j, int& l, int& r, int& c) {
    if (j < 128)       { l = 0; int loc = j;        r = loc / 8;  c = (loc - r * 8)  * 4; }
    else if (j < 512)  { l = 1; int loc = j - 128;  r = loc / 24; c = (loc - r * 24) * 4; }
    else if (j < 1152) { l = 2; int loc = j - 512;  r = loc / 40; c = (loc - r * 40) * 4; }
    else               { l = 3; int loc = j - 1152; r = loc / 56; c = (loc - r * 56) * 4; }
}

// Issue one tile's staging: exactly 8 b128 chunks per thread (uniform -> ASYNCcnt += 8 per wave)
__device__ __forceinline__ void issue_tile(const float* const* xs, long nodeBase, int N,
                                           float* buf, int tid) {
    const int mlA[4] = {1, 3, 5, 7};
    const int cbA[4] = {0, 32, 128, 288};
#pragma unroll
    for (int q = 0; q < 8; ++q) {
        int l, r, c;
        map_chunk(tid + q * NTHREADS, l, r, c);
        long n = nodeBase + r;
        if (n >= N) n = N - 1;                 // clamp: keeps per-wave issue count uniform
        const float* g = xs[l] + n * (long)(32 * mlA[l]) + c;
        issue_chunk(g, &buf[r * XSTRIDE + cbA[l] + c]);
    }
}

__global__ __launch_bounds__(NTHREADS)
void equi_linear_wmma(const float* __restrict__ x0, const float* __restrict__ x1,
                      const float* __restrict__ x2, const float* __restrict__ x3,
                      const float* __restrict__ w0, const float* __restrict__ w1,
                      const float* __restrict__ w2, const float* __restrict__ w3,
                      float* __restrict__ out, int N, int nTiles)
{
    __shared__ float ldsX[2][TILE * XSTRIDE];   // double-buffered x tile / output staging
    __shared__ float ldsW[16 * WSSTRIDE];       // W as [slice][k][o], k-stride padded

    const int tid    = threadIdx.x;
    const int wave   = tid >> 5;
    const int lane   = tid & 31;
    const int laneLo = lane & 15;
    const int half   = lane >> 4;

    const float* xs[4] = {x0, x1, x2, x3};
    const float* wg[4] = {w0, w1, w2, w3};
    const int mlArr[4] = {1, 3, 5, 7};
    const int sbArr[4] = {0, 1, 4, 9};

    // ---- stage W once: W_l[o][k][m] -> ldsW[slice][k][o] (64KB, L2-resident across blocks) ----
    for (int l = 0; l < 4; ++l) {
        const int ml  = mlArr[l];
        const int tot = 32 * 32 * ml;
        const float* src = wg[l];
        for (int i = tid; i < tot; i += NTHREADS) {
            const int o   = i / (32 * ml);
            const int rem = i - o * 32 * ml;
            const int k   = rem / ml;
            const int m   = rem - k * ml;
            ldsW[(sbArr[l] + m) * WSSTRIDE + k * WKSTRIDE + o] = src[i];
        }
    }

    // per-slice tables (slice = global (l,m) index 0..15)
    const int ML[16]  = {1,3,3,3,5,5,5,5,5,7,7,7,7,7,7,7};
    const int MM[16]  = {0,0,1,2,0,1,2,3,4,0,1,2,3,4,5,6};
    const int CBS[16] = {0,32,32,32,128,128,128,128,128,288,288,288,288,288,288,288};

    const int t0     = blockIdx.x;
    const int stride = gridDim.x;

    if (t0 < nTiles) issue_tile(xs, (long)t0 * TILE, N, ldsX[0], tid);

    int idx = 0;
    for (int t = t0; t < nTiles; t += stride, ++idx) {
        float* cur = ldsX[idx & 1];
        float* nxt = ldsX[(idx & 1) ^ 1];
        const long nodeBase = (long)t * TILE;
        const bool pref = (t + stride) < nTiles;          // block-uniform

        if (pref) {
            issue_tile(xs, (long)(t + stride) * TILE, N, nxt, tid);
            wait_async_le<8>();                           // in-order: current tile's 8 are done
        } else {
            wait_async_le<0>();
        }
        __syncthreads();

        // ---- compute: wave owns slices 2w,2w+1; 2 out-subtiles each; K=32 via 8x wmma 16x16x4 ----
        v8f acc[2][2];
#pragma unroll
        for (int si = 0; si < 2; ++si) {
            const int s  = wave * 2 + si;
            const int ml = ML[s];
            const int m  = MM[s];
            const int cb = CBS[s];
            const float* wBase = &ldsW[s * WSSTRIDE];
            const int arow = laneLo * XSTRIDE + cb + m;
#pragma unroll
            for (int ot = 0; ot < 2; ++ot) {
                v8f c = {0.f, 0.f, 0.f, 0.f, 0.f, 0.f, 0.f, 0.f};
#pragma unroll
                for (int kk = 0; kk < 8; ++kk) {
                    const int k0 = kk * 4 + half * 2;     // frag: VGPR v <-> K = k0 + v
                    v2f a, b;
                    a.x = cur[arow + k0 * ml];
                    a.y = cur[arow + (k0 + 1) * ml];
                    b.x = wBase[k0 * WKSTRIDE + ot * 16 + laneLo];
                    b.y = wBase[(k0 + 1) * WKSTRIDE + ot * 16 + laneLo];
                    c = __builtin_amdgcn_wmma_f32_16x16x4_f32(
                            false, a, false, b, (short)0, c, false, false);
                }
                acc[si][ot] = c;
            }
        }
        __syncthreads();   // all waves done reading cur -> reuse as output staging

        // ---- stage D tiles into cur (D layout: M = v + 8*(lane>>4), N = lane&15) ----
#pragma unroll
        for (int si = 0; si < 2; ++si) {
            const int s  = wave * 2 + si;
            const int ml = ML[s];
            const int m  = MM[s];
            const int cb = CBS[s];
#pragma unroll
            for (int ot = 0; ot < 2; ++ot) {
                const int col = cb + (ot * 16 + laneLo) * ml + m;
#pragma unroll
                for (int v = 0; v < 8; ++v) {
                    const int row = v + half * 8;
                    cur[row * XSTRIDE + col] = acc[si][ot][v];
                }
            }
        }
        __syncthreads();

        // ---- coalesced b128 stores of the staged output tile ----
        for (int i = tid; i < (TILE * ROW) >> 2; i += NTHREADS) {
            const int f = i << 2;
            const int r = f >> 9;
            const int c = f & (ROW - 1);
            if (nodeBase + r < N) {
                v4f v = *(const v4f*)(&cur[r * XSTRIDE + c]);
                *(v4f*)(out + (nodeBase + r) * (long)ROW + c) = v;
            }
        }
        __syncthreads();   // protect cur before async engine refills it in the next iteration
    }
}

extern "C" void kernel_launch(void* const* d_in, const int* in_sizes, int n_in,
                              void* d_out, int out_size, void* d_ws, size_t ws_size,
                              hipStream_t stream) {
    // setup_inputs() dict order: x0, W0, x1, W1, x2, W2, x3, W3
    const float* x0 = (const float*)d_in[0];
    const float* W0 = (const float*)d_in[1];
    const float* x1 = (const float*)d_in[2];
    const float* W1 = (const float*)d_in[3];
    const float* x2 = (const float*)d_in[4];
    const float* W2 = (const float*)d_in[5];
    const float* x3 = (const float*)d_in[6];
    const float* W3 = (const float*)d_in[7];
    float* out = (float*)d_out;

    const int N      = in_sizes[0] / 32;          // x0 is (N, 32, 1)
    const int nTiles = (N + TILE - 1) / TILE;     // 16384 for N = 262144
    const int grid   = nTiles < NBLOCKS ? nTiles : NBLOCKS;

    equi_linear_wmma<<<dim3(grid), dim3(NTHREADS), 0, stream>>>(
        x0, x1, x2, x3, W0, W1, W2, W3, out, N, nTiles);
}